// TopKGate_5385888989890
// MI455X (gfx1250) — compile-verified
//
#include <hip/hip_runtime.h>
#include <hip/hip_bf16.h>
#include <math.h>
#include <stdint.h>

#define S_TOK 4096
#define MDIM  1024
#define NEXP  8
#define CAP   1024

#define BT  128   // output tile (rows=slots, cols=N)
#define KT  16    // K step (double buffered)
#define AST 20    // A LDS row stride (floats): 80B rows -> 16B-aligned b128, distinct banks
#define BST 132   // B LDS row stride (floats): 528B rows -> 16B-aligned b128, (4k+col)%64 banks

typedef __attribute__((ext_vector_type(2))) float v2f;
typedef __attribute__((ext_vector_type(4))) float v4f;
typedef __attribute__((ext_vector_type(8))) float v8f;

// Async global -> LDS 16-byte copy (per lane), tracked by ASYNCcnt.
__device__ __forceinline__ void async_copy16(void* lds, const void* gaddr) {
    unsigned loff = (unsigned)(uintptr_t)lds;   // low 32 bits of generic ptr = LDS offset
    asm volatile("global_load_async_to_lds_b128 %0, %1, off"
                 :: "v"(loff), "v"(gaddr) : "memory");
}
__device__ __forceinline__ void async_wait_all() {
    asm volatile("s_wait_asynccnt 0x0" ::: "memory");
}

// ---------------------------------------------------------------------------
// Kernel 1: gating. One wave32 per token. logits = x[s] . wg[e], softmax,
// argmax (lowest index on tie, matching lax.top_k).
// ---------------------------------------------------------------------------
__global__ __launch_bounds__(256)
void gate_kernel(const float* __restrict__ x,
                 const float* __restrict__ wg,
                 float* __restrict__ gates8,   // [S][8]
                 float* __restrict__ gate0,    // [S]
                 int*   __restrict__ idx0)     // [S]
{
    const int wave = (blockIdx.x * blockDim.x + threadIdx.x) >> 5; // token
    const int lane = threadIdx.x & 31;
    const float* xr = x + (size_t)wave * MDIM;

    float acc[NEXP];
#pragma unroll
    for (int e = 0; e < NEXP; ++e) acc[e] = 0.f;

    for (int m = lane; m < MDIM; m += 32) {
        float xv = xr[m];
#pragma unroll
        for (int e = 0; e < NEXP; ++e)
            acc[e] += xv * wg[e * MDIM + m];
    }
#pragma unroll
    for (int e = 0; e < NEXP; ++e) {
#pragma unroll
        for (int off = 16; off > 0; off >>= 1)
            acc[e] += __shfl_xor(acc[e], off, 32);
    }
    if (lane == 0) {
        float mx = acc[0]; int im = 0;
#pragma unroll
        for (int e = 1; e < NEXP; ++e)
            if (acc[e] > mx) { mx = acc[e]; im = e; }
        float ex[NEXP]; float den = 0.f;
#pragma unroll
        for (int e = 0; e < NEXP; ++e) { ex[e] = expf(acc[e] - mx); den += ex[e]; }
        const float inv = 1.0f / den;
#pragma unroll
        for (int e = 0; e < NEXP; ++e) gates8[wave * NEXP + e] = ex[e] * inv;
        gate0[wave] = ex[im] * inv;
        idx0[wave]  = im;
    }
}

// ---------------------------------------------------------------------------
// Kernel 2: deterministic per-expert stable scan (cumsum of argmax one-hot).
// Single wave32: ballot + popcount segmented scan, 128 steps.
// tok[e*CAP + loc] = s for tokens that win a capacity slot (-1 elsewhere,
// pre-initialized via memset 0xFF).
// ---------------------------------------------------------------------------
__global__ void scan_kernel(const int* __restrict__ idx0,
                            int* __restrict__ tok)
{
    const int lane = threadIdx.x;
    int cnt[NEXP];
#pragma unroll
    for (int e = 0; e < NEXP; ++e) cnt[e] = 0;

    const unsigned int lowmask = (lane == 0) ? 0u : (0xFFFFFFFFu >> (32 - lane));

    for (int base = 0; base < S_TOK; base += 32) {
        const int s = base + lane;
        const int e = idx0[s];
#pragma unroll
        for (int em = 0; em < NEXP; ++em) {
            const unsigned int b32 = (unsigned int)__ballot(e == em);
            if (e == em) {
                const int loc = cnt[em] + __popc(b32 & lowmask);
                if (loc < CAP) tok[em * CAP + loc] = s;
            }
            cnt[em] += __popc(b32);
        }
    }
}

// ---------------------------------------------------------------------------
// Kernel 3: per-expert GEMM, fp32 WMMA 16x16x4, fused gather (A rows =
// x[token_of_slot], zero rows for empty slots) and fused scatter epilogue
// (out[token] = gate * row). Double-buffered LDS tiles filled with
// GLOBAL_LOAD_ASYNC_TO_LDS_B128 so step k+1 streams in while step k computes.
// Grid: (N/128, CAP/128, E), block 256 = 8 waves; each wave owns a 64x32
// region = 4x2 grid of 16x16 accumulators.
// ---------------------------------------------------------------------------
__global__ __launch_bounds__(256)
void moe_gemm_kernel(const float* __restrict__ x,
                     const float* __restrict__ we,
                     const int*   __restrict__ tok,
                     const float* __restrict__ gate0,
                     float* __restrict__ out)
{
    __shared__ float As[2][BT * AST];
    __shared__ float Bs[2][KT * BST];
    __shared__ int   tokRow[BT];
    __shared__ float gateRow[BT];

    const int nt   = blockIdx.x;       // N tile
    const int ct   = blockIdx.y;       // slot tile
    const int e    = blockIdx.z;       // expert
    const int t    = threadIdx.x;
    const int lane = t & 31;
    const int wave = t >> 5;
    const int half = lane >> 4;        // 0: lanes 0-15, 1: lanes 16-31
    const int m16  = lane & 15;
    const int wc   = wave >> 2;        // 0..1  (row direction, 64 rows each)
    const int wn   = wave & 3;         // 0..3  (col direction, 32 cols each)

    if (t < BT) {
        const int s = tok[e * CAP + ct * BT + t];
        tokRow[t]  = s;
        gateRow[t] = (s >= 0) ? gate0[s] : 0.f;
    }
    __syncthreads();

    v8f acc[4][2];
#pragma unroll
    for (int i = 0; i < 4; ++i)
#pragma unroll
        for (int j = 0; j < 2; ++j)
#pragma unroll
            for (int r = 0; r < 8; ++r) acc[i][j][r] = 0.f;

    // A loader: 2 threads per row, 8 contiguous floats each
    const int ar  = t >> 1;
    const int acs = (t & 1) * 8;
    const int sA  = tokRow[ar];
    const float* xrow = (sA >= 0) ? (x + (size_t)sA * MDIM) : (const float*)0;
    // B loader: 16 threads per row, 8 contiguous floats each
    const int br  = t >> 4;
    const int bcs = (t & 15) * 8;
    const float* wbase = we + (size_t)e * MDIM * MDIM + (size_t)nt * BT;

    // Empty-slot rows never get async writes: zero them once in both buffers.
    if (!xrow) {
        v4f z; z.x = 0.f; z.y = 0.f; z.z = 0.f; z.w = 0.f;
#pragma unroll
        for (int b = 0; b < 2; ++b)
#pragma unroll
            for (int j = 0; j < 2; ++j)
                *(v4f*)(&As[b][ar * AST + acs + j * 4]) = z;
    }

    // Issue async fill of one K-slice into buffer `buf`.
    auto load_tile = [&](int k0, int buf) {
        if (xrow) {
#pragma unroll
            for (int j = 0; j < 2; ++j)
                async_copy16(&As[buf][ar * AST + acs + j * 4],
                             xrow + k0 + acs + j * 4);
        }
#pragma unroll
        for (int j = 0; j < 2; ++j)
            async_copy16(&Bs[buf][br * BST + bcs + j * 4],
                         wbase + (size_t)(k0 + br) * MDIM + bcs + j * 4);
    };

    load_tile(0, 0);
    async_wait_all();
    __syncthreads();

    int buf = 0;
    for (int k0 = 0; k0 < MDIM; k0 += KT) {
        if (k0 + KT < MDIM) load_tile(k0 + KT, buf ^ 1);   // prefetch next slice

#pragma unroll
        for (int kk = 0; kk < KT; kk += 4) {
            v2f afrag[4];
            v2f bfrag[2];
#pragma unroll
            for (int mt = 0; mt < 4; ++mt) {
                const int row = wc * 64 + mt * 16 + m16;
                afrag[mt] = *(const v2f*)(&As[buf][row * AST + kk + half * 2]);
            }
#pragma unroll
            for (int nn = 0; nn < 2; ++nn) {
                const int col = wn * 32 + nn * 16 + m16;
                bfrag[nn].x = Bs[buf][(kk + half * 2 + 0) * BST + col];
                bfrag[nn].y = Bs[buf][(kk + half * 2 + 1) * BST + col];
            }
#pragma unroll
            for (int mt = 0; mt < 4; ++mt)
#pragma unroll
                for (int nn = 0; nn < 2; ++nn)
                    acc[mt][nn] = __builtin_amdgcn_wmma_f32_16x16x4_f32(
                        false, afrag[mt], false, bfrag[nn],
                        (short)0, acc[mt][nn], false, false);
        }

        async_wait_all();   // prefetched slice landed in LDS
        __syncthreads();
        buf ^= 1;
    }

    // Epilogue: scatter rows to their tokens, scaled by gate.
#pragma unroll
    for (int mt = 0; mt < 4; ++mt) {
#pragma unroll
        for (int nn = 0; nn < 2; ++nn) {
#pragma unroll
            for (int r = 0; r < 8; ++r) {
                const int mloc = wc * 64 + mt * 16 + half * 8 + r;
                const int s = tokRow[mloc];
                if (s >= 0) {
                    const int col = nt * BT + wn * 32 + nn * 16 + m16;
                    out[(size_t)s * MDIM + col] = gateRow[mloc] * acc[mt][nn][r];
                }
            }
        }
    }
}

// ---------------------------------------------------------------------------
// Kernel 4: load-balance loss, deterministic fixed-order reduction.
// l = sum_e me[e]*ce[e] * E/S^2
// ---------------------------------------------------------------------------
__global__ __launch_bounds__(256)
void loss_kernel(const float* __restrict__ gates8,
                 const int*   __restrict__ idx0,
                 float* __restrict__ out_loss)
{
    __shared__ float sme[256 * NEXP];
    __shared__ float sce[256 * NEXP];
    const int t = threadIdx.x;
    float me[NEXP], ce[NEXP];
#pragma unroll
    for (int e = 0; e < NEXP; ++e) { me[e] = 0.f; ce[e] = 0.f; }
    for (int s = t; s < S_TOK; s += 256) {
        ce[idx0[s]] += 1.f;
#pragma unroll
        for (int e = 0; e < NEXP; ++e) me[e] += gates8[s * NEXP + e];
    }
#pragma unroll
    for (int e = 0; e < NEXP; ++e) { sme[t * NEXP + e] = me[e]; sce[t * NEXP + e] = ce[e]; }
    __syncthreads();
    for (int stride = 128; stride > 0; stride >>= 1) {
        if (t < stride) {
#pragma unroll
            for (int e = 0; e < NEXP; ++e) {
                sme[t * NEXP + e] += sme[(t + stride) * NEXP + e];
                sce[t * NEXP + e] += sce[(t + stride) * NEXP + e];
            }
        }
        __syncthreads();
    }
    if (t == 0) {
        float l = 0.f;
#pragma unroll
        for (int e = 0; e < NEXP; ++e) l += sme[e] * sce[e];
        out_loss[0] = l * ((float)NEXP / ((float)S_TOK * (float)S_TOK));
    }
}

// ---------------------------------------------------------------------------
extern "C" void kernel_launch(void* const* d_in, const int* in_sizes, int n_in,
                              void* d_out, int out_size, void* d_ws, size_t ws_size,
                              hipStream_t stream) {
    const float* x  = (const float*)d_in[0];   // [S, M]
    const float* wg = (const float*)d_in[1];   // [E, M]
    const float* we = (const float*)d_in[2];   // [E, M, M]
    float* out = (float*)d_out;                // [S*M] result + [1] loss

    // workspace layout (192 KB total)
    char* ws = (char*)d_ws;
    float* gates8 = (float*)ws;                       // S*8 floats
    float* gate0  = gates8 + (size_t)S_TOK * NEXP;    // S floats
    int*   idx0   = (int*)(gate0 + S_TOK);            // S ints
    int*   tok    = idx0 + S_TOK;                     // E*CAP ints

    hipMemsetAsync(out, 0, (size_t)out_size * sizeof(float), stream);
    hipMemsetAsync(tok, 0xFF, (size_t)NEXP * CAP * sizeof(int), stream); // -1

    gate_kernel<<<S_TOK / 8, 256, 0, stream>>>(x, wg, gates8, gate0, idx0);
    scan_kernel<<<1, 32, 0, stream>>>(idx0, tok);

    dim3 grid(MDIM / BT, CAP / BT, NEXP);
    moe_gemm_kernel<<<grid, 256, 0, stream>>>(x, we, tok, gate0, out);

    loss_kernel<<<1, 256, 0, stream>>>(gates8, idx0, out + (size_t)S_TOK * MDIM);
}